// QuantumKernelMethod_65481071401003
// MI455X (gfx1250) — compile-verified
//
#include <hip/hip_runtime.h>
#include <hip/hip_bf16.h>

typedef __attribute__((ext_vector_type(2))) float v2f;
typedef __attribute__((ext_vector_type(8))) float v8f;

#define GAMMA   0.00390625f
#define DIM     256
#define NTRAIN  8192
#define NTEST   4096
#define RT      128         // gram block tile rows
#define CT      64          // gram block tile cols
#define KC      64          // k-chunk staged in LDS
#define PITCH   68          // LDS row pitch (floats), avoids bank conflicts
#define NITER   256         // CG iterations (fixed -> deterministic)

// ---------------------------------------------------------------------------
// Wave reduction helper (wave32)
// ---------------------------------------------------------------------------
__device__ __forceinline__ float wave_reduce_add(float s) {
#pragma unroll
    for (int off = 16; off > 0; off >>= 1)
        s += __shfl_xor(s, off, 32);
    return s;
}

// ---------------------------------------------------------------------------
// Row squared-norms: one wave per row, 256 floats per row
// ---------------------------------------------------------------------------
__global__ __launch_bounds__(256)
void rownorm_kernel(const float* __restrict__ X, float* __restrict__ sq, int rows) {
    const int row  = blockIdx.x * 8 + (threadIdx.x >> 5);
    const int lane = threadIdx.x & 31;
    if (row >= rows) return;
    const float4* X4 = (const float4*)(X + (size_t)row * DIM);
    float s = 0.f;
#pragma unroll
    for (int c = lane; c < DIM / 4; c += 32) {
        float4 a = X4[c];
        s += a.x * a.x + a.y * a.y + a.z * a.z + a.w * a.w;
    }
    s = wave_reduce_add(s);
    if (lane == 0) sq[row] = s;
}

// ---------------------------------------------------------------------------
// RBF Gram tile kernel using V_WMMA_F32_16X16X4_F32 (exact f32 matmul).
// Block = 128 threads (4 waves) computes a 128x64 tile of
//   K[i][j] = exp(-GAMMA * max(0, sqA[i] + sqB[j] - 2 * <Xa_i, Xb_j>)) (+1 if i==j)
// Wave w computes rows [32w, 32w+32) x 64 cols: 2 row sub-tiles x 4 col
// sub-tiles = 8 accumulators; B fragments reused across both row strips.
// A-operand (16x4 f32): lanes 0-15 -> M=lane, K=k+{0,1};
//                       lanes 16-31 -> M=lane-16, K=k+{2,3}.
// B-operand (4x16 f32): lanes 0-15 -> N=lane, K=k+{0,1};
//                       lanes 16-31 -> N=lane-16, K=k+{2,3}.
// C layout: VGPR v, lanes 0-15: (M=v, N=lane); lanes 16-31: (M=v+8, N=lane-16).
// ---------------------------------------------------------------------------
__global__ __launch_bounds__(128)
void rbf_gram_wmma(const float* __restrict__ Xa, const float* __restrict__ Xb,
                   const float* __restrict__ sqA, const float* __restrict__ sqB,
                   float* __restrict__ K, int na, int nb, int addDiag) {
    __shared__ float As[RT * PITCH];
    __shared__ float Bs[CT * PITCH];

    const int i0   = blockIdx.y * RT;
    const int j0   = blockIdx.x * CT;
    const int t    = threadIdx.x;
    const int wave = t >> 5;
    const int lane = t & 31;
    const int l15  = lane & 15;
    const int koff = (lane >> 4) << 1;   // 0 or 2

    v8f acc[2][4];
    const v8f vzero = {};
#pragma unroll
    for (int r = 0; r < 2; ++r)
#pragma unroll
        for (int s = 0; s < 4; ++s) acc[r][s] = vzero;

    for (int k0 = 0; k0 < DIM; k0 += KC) {
        // Stage A: 128 rows x 64 floats (2048 float4, 16 per thread)
#pragma unroll
        for (int it = 0; it < 16; ++it) {
            const int f   = t + 128 * it;
            const int row = f >> 4;
            const int c4  = f & 15;
            float4 a = *(const float4*)(Xa + (size_t)(i0 + row) * DIM + k0 + c4 * 4);
            *(float4*)(&As[row * PITCH + c4 * 4]) = a;
        }
        // Stage B: 64 rows x 64 floats (1024 float4, 8 per thread)
#pragma unroll
        for (int it = 0; it < 8; ++it) {
            const int f   = t + 128 * it;
            const int row = f >> 4;
            const int c4  = f & 15;
            float4 b = *(const float4*)(Xb + (size_t)(j0 + row) * DIM + k0 + c4 * 4);
            *(float4*)(&Bs[row * PITCH + c4 * 4]) = b;
        }
        __syncthreads();

        const int arow0 = (wave << 5) | l15;          // row sub-tile 0
#pragma unroll
        for (int kk = 0; kk < KC; kk += 4) {
            const v2f a0 = *(const v2f*)(&As[arow0 * PITCH + kk + koff]);
            const v2f a1 = *(const v2f*)(&As[(arow0 + 16) * PITCH + kk + koff]);
#pragma unroll
            for (int s = 0; s < 4; ++s) {
                const int brow = (s << 4) | l15;
                const v2f bf = *(const v2f*)(&Bs[brow * PITCH + kk + koff]);
                acc[0][s] = __builtin_amdgcn_wmma_f32_16x16x4_f32(
                    false, a0, false, bf, (short)0, acc[0][s], false, false);
                acc[1][s] = __builtin_amdgcn_wmma_f32_16x16x4_f32(
                    false, a1, false, bf, (short)0, acc[1][s], false, false);
            }
        }
        __syncthreads();
    }

    // Epilogue: dot -> RBF kernel value.
#pragma unroll
    for (int r = 0; r < 2; ++r) {
        const int mbase = i0 + (wave << 5) + (r << 4) + ((lane >> 4) << 3);
#pragma unroll
        for (int s = 0; s < 4; ++s) {
            const int gn  = j0 + (s << 4) + l15;
            const float b = sqB[gn];
#pragma unroll
            for (int v = 0; v < 8; ++v) {
                const int gm = mbase + v;
                float d2 = sqA[gm] + b - 2.0f * acc[r][s][v];
                d2 = fmaxf(d2, 0.0f);
                float kv = __expf(-GAMMA * d2);
                if (addDiag && gm == gn) kv += 1.0f;   // ridge alpha = 1.0
                K[(size_t)gm * nb + gn] = kv;
            }
        }
    }
}

// ---------------------------------------------------------------------------
// Dense matvec: out[row] = dot(K[row][:], v).  One wave per row, float4 loads.
// ---------------------------------------------------------------------------
__global__ __launch_bounds__(256)
void matvec_kernel(const float* __restrict__ K, const float* __restrict__ v,
                   float* __restrict__ out, int rows, int cols) {
    const int row  = blockIdx.x * 8 + (threadIdx.x >> 5);
    const int lane = threadIdx.x & 31;
    if (row >= rows) return;
    const float4* K4 = (const float4*)(K + (size_t)row * cols);
    const float4* v4 = (const float4*)v;
    float s = 0.f;
    for (int c = lane; c < cols / 4; c += 32) {
        float4 a = K4[c], b = v4[c];
        s += a.x * b.x + a.y * b.y + a.z * b.z + a.w * b.w;
    }
    s = wave_reduce_add(s);
    if (lane == 0) out[row] = s;
}

// ---------------------------------------------------------------------------
// Deterministic dot product: single block, fixed reduction tree.
// ---------------------------------------------------------------------------
__global__ __launch_bounds__(1024)
void dot_kernel(const float* __restrict__ a, const float* __restrict__ b,
                float* __restrict__ out, int n) {
    __shared__ float sm[32];
    const int t = threadIdx.x;
    float s = 0.f;
    for (int i = t; i < n; i += 1024) s += a[i] * b[i];
    s = wave_reduce_add(s);
    if ((t & 31) == 0) sm[t >> 5] = s;
    __syncthreads();
    if (t < 32) {
        float w = sm[t];
        w = wave_reduce_add(w);
        if (t == 0) *out = w;
    }
}

// ---------------------------------------------------------------------------
// CG bookkeeping (device-side scalars; no host sync -> graph-capturable)
// scal[0]=rs_old  scal[1]=pAp  scal[2]=alpha  scal[3]=beta  scal[4]=rs_new
// ---------------------------------------------------------------------------
__global__ void cg_init_kernel(float* __restrict__ x, float* __restrict__ r,
                               float* __restrict__ p, const float* __restrict__ y, int n) {
    const int i = blockIdx.x * 256 + threadIdx.x;
    if (i < n) { x[i] = 0.f; float yi = y[i]; r[i] = yi; p[i] = yi; }
}

__global__ void cg_alpha_kernel(float* __restrict__ scal) {
    const float pAp = scal[1];
    scal[2] = (pAp != 0.f) ? (scal[0] / pAp) : 0.f;
}

__global__ void cg_update_xr_kernel(float* __restrict__ x, float* __restrict__ r,
                                    const float* __restrict__ p, const float* __restrict__ Ap,
                                    const float* __restrict__ scal, int n) {
    const int i = blockIdx.x * 256 + threadIdx.x;
    if (i < n) { const float a = scal[2]; x[i] += a * p[i]; r[i] -= a * Ap[i]; }
}

__global__ void cg_beta_kernel(float* __restrict__ scal) {
    const float rso = scal[0], rsn = scal[4];
    scal[3] = (rso != 0.f) ? (rsn / rso) : 0.f;
    scal[0] = rsn;
}

__global__ void cg_update_p_kernel(float* __restrict__ p, const float* __restrict__ r,
                                   const float* __restrict__ scal, int n) {
    const int i = blockIdx.x * 256 + threadIdx.x;
    if (i < n) p[i] = r[i] + scal[3] * p[i];
}

// ---------------------------------------------------------------------------
// Host-side orchestration (all on `stream`, no sync, graph-capture safe)
// ---------------------------------------------------------------------------
extern "C" void kernel_launch(void* const* d_in, const int* in_sizes, int n_in,
                              void* d_out, int out_size, void* d_ws, size_t ws_size,
                              hipStream_t stream) {
    (void)in_sizes; (void)n_in; (void)out_size; (void)ws_size;

    const float* Xtr = (const float*)d_in[0];   // [8192, 256]
    const float* Xte = (const float*)d_in[1];   // [4096, 256]
    const float* ytr = (const float*)d_in[2];   // [8192]
    float* out = (float*)d_out;                 // [4096]

    float* w    = (float*)d_ws;
    float* K    = w;                                    // 8192*8192
    float* Kt   = K  + (size_t)NTRAIN * NTRAIN;         // 4096*8192
    float* sqA  = Kt + (size_t)NTEST * NTRAIN;          // 8192
    float* sqB  = sqA + NTRAIN;                         // 4096
    float* x    = sqB + NTEST;                          // 8192 (coef)
    float* r    = x + NTRAIN;
    float* p    = r + NTRAIN;
    float* Ap   = p + NTRAIN;
    float* scal = Ap + NTRAIN;                          // 8 floats

    // 1) row squared norms
    rownorm_kernel<<<NTRAIN / 8, 256, 0, stream>>>(Xtr, sqA, NTRAIN);
    rownorm_kernel<<<NTEST  / 8, 256, 0, stream>>>(Xte, sqB, NTEST);

    // 2) Gram matrices via f32 WMMA
    {
        dim3 g(NTRAIN / CT, NTRAIN / RT);
        rbf_gram_wmma<<<g, 128, 0, stream>>>(Xtr, Xtr, sqA, sqA, K, NTRAIN, NTRAIN, 1);
    }
    {
        dim3 g(NTRAIN / CT, NTEST / RT);
        rbf_gram_wmma<<<g, 128, 0, stream>>>(Xte, Xtr, sqB, sqA, Kt, NTEST, NTRAIN, 0);
    }

    // 3) CG solve of (K + I) x = y   (ridge already baked into K diagonal)
    cg_init_kernel<<<NTRAIN / 256, 256, 0, stream>>>(x, r, p, ytr, NTRAIN);
    dot_kernel<<<1, 1024, 0, stream>>>(r, r, &scal[0], NTRAIN);   // rs_old

    for (int it = 0; it < NITER; ++it) {
        matvec_kernel<<<NTRAIN / 8, 256, 0, stream>>>(K, p, Ap, NTRAIN, NTRAIN);
        dot_kernel<<<1, 1024, 0, stream>>>(p, Ap, &scal[1], NTRAIN);      // pAp
        cg_alpha_kernel<<<1, 1, 0, stream>>>(scal);
        cg_update_xr_kernel<<<NTRAIN / 256, 256, 0, stream>>>(x, r, p, Ap, scal, NTRAIN);
        dot_kernel<<<1, 1024, 0, stream>>>(r, r, &scal[4], NTRAIN);       // rs_new
        cg_beta_kernel<<<1, 1, 0, stream>>>(scal);
        cg_update_p_kernel<<<NTRAIN / 256, 256, 0, stream>>>(p, r, scal, NTRAIN);
    }

    // 4) y_pred = K_test @ coef
    matvec_kernel<<<NTEST / 8, 256, 0, stream>>>(Kt, x, out, NTEST, NTRAIN);
}